// MutiHeadAttention_16389595201938
// MI455X (gfx1250) — compile-verified
//
#include <hip/hip_runtime.h>

// Problem constants (match reference)
#define B_   2
#define QL_  2048
#define KL_  2048
#define DM_  1024
#define NH_  16
#define DK_  64

typedef float v2f __attribute__((ext_vector_type(2)));
typedef float v8f __attribute__((ext_vector_type(8)));

// f32 WMMA: D(16x16,f32) = A(16x4,f32) * B(4x16,f32) + C
// Per-lane operands (ISA 7.12.2): lanes 0-15 hold K={0,1}, lanes 16-31 hold K={2,3}.
__device__ __forceinline__ v8f wmma_f32(v2f a, v2f b, v8f c) {
  return __builtin_amdgcn_wmma_f32_16x16x4_f32(false, a, false, b, (short)0, c,
                                               false, false);
}

// ---------------------------------------------------------------------------
// C[M,N] = A[M,K] @ W[N,K]^T  (+ optional residual R).
// One wave -> one 16(M) x 64(N) output strip: 4 v8f accumulators; the A
// fragment is loaded once per k-step and reused across 4 B fragments.
// The k-loop is software-pipelined: next step's 5 fragment loads are issued
// into fresh registers BEFORE the current step's 4 WMMAs, so the WMMAs never
// wait on a just-issued load (avoids the s_wait_loadcnt 0x0 serialization the
// compiler produced otherwise).
// ---------------------------------------------------------------------------
__global__ void __launch_bounds__(128)
gemm16_nt_kernel(const float* __restrict__ A, const float* __restrict__ W,
                 const float* __restrict__ R, float* __restrict__ C,
                 int M, int N, int K, int add_res) {
  int wave = blockIdx.x * (blockDim.x >> 5) + (threadIdx.x >> 5);
  int lane = threadIdx.x & 31;
  int nstrips = N >> 6;                       // 64-wide N strips
  int mt = wave / nstrips;
  int ns = wave - mt * nstrips;
  if (mt >= (M >> 4)) return;                 // wave-uniform guard (EXEC full)

  int l15 = lane & 15;
  int sel = (lane >> 4) << 1;                 // 0 for lanes 0-15, 2 for 16-31
  int hi8 = (lane >> 4) << 3;

  const float* arow = A + (size_t)(mt * 16 + l15) * K + sel;
  const float* w0 = W + (size_t)(ns * 64 + 0  + l15) * K + sel;
  const float* w1 = W + (size_t)(ns * 64 + 16 + l15) * K + sel;
  const float* w2 = W + (size_t)(ns * 64 + 32 + l15) * K + sel;
  const float* w3 = W + (size_t)(ns * 64 + 48 + l15) * K + sel;

  // Pipeline prologue: fragments for k = 0.
  v2f a  = *(const v2f*)(arow);
  v2f b0 = *(const v2f*)(w0);
  v2f b1 = *(const v2f*)(w1);
  v2f b2 = *(const v2f*)(w2);
  v2f b3 = *(const v2f*)(w3);

  v8f acc0 = {}, acc1 = {}, acc2 = {}, acc3 = {};
  for (int k = 0; k < K; k += 4) {
    // Prefetch next k-step (last iteration re-reads k=0: in-bounds, unused).
    int kn = (k + 4 < K) ? (k + 4) : 0;
    v2f an  = *(const v2f*)(arow + kn);
    v2f bn0 = *(const v2f*)(w0 + kn);
    v2f bn1 = *(const v2f*)(w1 + kn);
    v2f bn2 = *(const v2f*)(w2 + kn);
    v2f bn3 = *(const v2f*)(w3 + kn);
    // Consume fragments loaded one iteration ago.
    acc0 = wmma_f32(a, b0, acc0);
    acc1 = wmma_f32(a, b1, acc1);
    acc2 = wmma_f32(a, b2, acc2);
    acc3 = wmma_f32(a, b3, acc3);
    a = an; b0 = bn0; b1 = bn1; b2 = bn2; b3 = bn3;
  }

  // C/D layout: VGPR r -> row r (lanes 0-15) / row r+8 (lanes 16-31), col = l15
  #pragma unroll
  for (int r = 0; r < 8; ++r) {
    int m = mt * 16 + r + hi8;
    size_t o = (size_t)m * N + ns * 64 + l15;
    float v0 = acc0[r], v1 = acc1[r], v2 = acc2[r], v3 = acc3[r];
    if (add_res) {
      v0 += R[o];
      v1 += R[o + 16];
      v2 += R[o + 32];
      v3 += R[o + 48];
    }
    C[o]      = v0;
    C[o + 16] = v1;
    C[o + 32] = v2;
    C[o + 48] = v3;
  }
}

// ---------------------------------------------------------------------------
// vsum[b, n*64+d] = sum_k vh[b, k, n*64+d]   (reference einsum sums the
// v-label independently of the attention weights -> only column sums of vh)
// ---------------------------------------------------------------------------
__global__ void vsum_kernel(const float* __restrict__ vh, float* __restrict__ vsum) {
  int i = blockIdx.x * blockDim.x + threadIdx.x;   // 0 .. B_*DM_-1
  if (i >= B_ * DM_) return;
  int b = i >> 10, c = i & (DM_ - 1);
  const float* p = vh + ((size_t)b << 21) + c;     // b * KL_ * DM_
  float s = 0.0f;
  for (int k = 0; k < KL_; ++k) s += p[(size_t)k << 10];
  vsum[i] = s;
}

// ---------------------------------------------------------------------------
// Fused attention. One wave per (b, head, 16-row q tile).
// Pass 1: scores via WMMA, online max/sum (half-wave shfl_xor reductions).
// Pass 2: recompute scores (L2-hot: kh is 16MB << 192MB L2), emit normalized
//         probabilities to `attn`, accumulate row sums, then
//         concat = rowsum(attn) * vsum through the reference reshape scramble.
// Mask applied BEFORE 1/sqrt(dk) scaling, exactly as in the reference.
// ---------------------------------------------------------------------------
__global__ void __launch_bounds__(32)
attn_kernel(const float* __restrict__ qh, const float* __restrict__ kh,
            const int* __restrict__ mask, const float* __restrict__ vsum,
            float* __restrict__ attn, float* __restrict__ concat) {
  int w    = blockIdx.x;                  // 0 .. B_*NH_*(QL_/16)-1
  int lane = threadIdx.x & 31;
  int b  = w >> 11;                       // / (NH_ * 128)
  int n  = (w >> 7) & (NH_ - 1);
  int qt = w & 127;
  int q0 = qt << 4;
  int l15 = lane & 15;
  int sel = (lane >> 4) << 1;
  int hi8 = (lane >> 4) << 3;

  // Preload the 16x64 Q tile as 16 A-fragments (float2 per lane per k-step).
  const float* qp = qh + ((size_t)(b * QL_ + q0 + l15) << 10) + n * DK_ + sel;
  v2f qa[16];
  #pragma unroll
  for (int j = 0; j < 16; ++j) qa[j] = *(const v2f*)(qp + 4 * j);

  const float* kbase = kh + ((size_t)(b * KL_) << 10) + n * DK_ + sel;
  const int*   mbase = mask + ((size_t)b << 22);       // b * QL_ * KL_

  float mrow[8], lrow[8];
  #pragma unroll
  for (int r = 0; r < 8; ++r) { mrow[r] = -3.0e38f; lrow[r] = 0.0f; }

  // ---- pass 1: online softmax statistics ----
  for (int kt = 0; kt < KL_ / 16; ++kt) {
    const float* kp = kbase + ((size_t)(kt * 16 + l15) << 10);
    v8f s = {};
    #pragma unroll
    for (int j = 0; j < 16; ++j) {
      v2f kb = *(const v2f*)(kp + 4 * j);              // B[d][kk'] = K[kk'][d]
      s = wmma_f32(qa[j], kb, s);
    }
    #pragma unroll
    for (int r = 0; r < 8; ++r) {
      int q  = q0 + r + hi8;
      int mk = mbase[(size_t)q * KL_ + kt * 16 + l15];
      float t = (mk != 0 ? s[r] : -1.0e9f) * 0.125f;   // mask, then /sqrt(64)
      float tm = t;                                     // max over the 16 lanes
      #pragma unroll
      for (int off = 1; off < 16; off <<= 1) tm = fmaxf(tm, __shfl_xor(tm, off, 32));
      float nm = fmaxf(mrow[r], tm);
      float e = __expf(t - nm);
      #pragma unroll
      for (int off = 1; off < 16; off <<= 1) e += __shfl_xor(e, off, 32);
      lrow[r] = lrow[r] * __expf(mrow[r] - nm) + e;
      mrow[r] = nm;
    }
  }

  float inv[8], rs[8];
  #pragma unroll
  for (int r = 0; r < 8; ++r) { inv[r] = 1.0f / lrow[r]; rs[r] = 0.0f; }

  // ---- pass 2: emit attn, accumulate row sums ----
  float* abase = attn + ((size_t)((b * NH_ + n) * QL_ + q0) << 11);
  for (int kt = 0; kt < KL_ / 16; ++kt) {
    const float* kp = kbase + ((size_t)(kt * 16 + l15) << 10);
    v8f s = {};
    #pragma unroll
    for (int j = 0; j < 16; ++j) {
      v2f kb = *(const v2f*)(kp + 4 * j);
      s = wmma_f32(qa[j], kb, s);
    }
    #pragma unroll
    for (int r = 0; r < 8; ++r) {
      int q  = q0 + r + hi8;
      int mk = mbase[(size_t)q * KL_ + kt * 16 + l15];
      float t = (mk != 0 ? s[r] : -1.0e9f) * 0.125f;
      float p = __expf(t - mrow[r]) * inv[r];
      abase[((size_t)(r + hi8) << 11) + kt * 16 + l15] = p;
      float ps = p;
      #pragma unroll
      for (int off = 1; off < 16; off <<= 1) ps += __shfl_xor(ps, off, 32);
      rs[r] += ps;
    }
  }

  // qkv[b,n,q,d] = rowsum(attn) * vsum[b,n,d]; reference reshape (no transpose):
  // concat[b][n*128 + q/16][(q%16)*64 + d]
  float* cb = concat + ((size_t)(b * QL_ + n * 128 + qt) << 10);
  const float* vs = vsum + b * DM_ + n * DK_;
  #pragma unroll
  for (int t = 0; t < 4; ++t) {
    float vv = vs[t * 16 + l15];
    #pragma unroll
    for (int r = 0; r < 8; ++r)
      cb[(r + hi8) * 64 + t * 16 + l15] = rs[r] * vv;
  }
}

// ---------------------------------------------------------------------------
// Row-wise LayerNorm over 1024 elements. One 256-thread block per row.
// ---------------------------------------------------------------------------
__global__ void __launch_bounds__(256)
ln_kernel(const float* __restrict__ x, const float* __restrict__ g,
          const float* __restrict__ bt, float* __restrict__ y) {
  __shared__ float red[12];
  int row = blockIdx.x, tid = threadIdx.x;
  const float* xr = x + ((size_t)row << 10);
  float4 v = *(const float4*)(xr + tid * 4);

  float s = v.x + v.y + v.z + v.w;
  #pragma unroll
  for (int off = 16; off >= 1; off >>= 1) s += __shfl_xor(s, off, 32);
  if ((tid & 31) == 0) red[tid >> 5] = s;
  __syncthreads();
  if (tid == 0) {
    float t = 0.0f;
    for (int i = 0; i < 8; ++i) t += red[i];
    red[8] = t * (1.0f / 1024.0f);
  }
  __syncthreads();
  float mu = red[8];

  float dx = v.x - mu, dy = v.y - mu, dz = v.z - mu, dw = v.w - mu;
  float q2 = dx * dx + dy * dy + dz * dz + dw * dw;
  #pragma unroll
  for (int off = 16; off >= 1; off >>= 1) q2 += __shfl_xor(q2, off, 32);
  if ((tid & 31) == 0) red[tid >> 5] = q2;
  __syncthreads();
  if (tid == 0) {
    float t = 0.0f;
    for (int i = 0; i < 8; ++i) t += red[i];
    red[9] = rsqrtf(t * (1.0f / 1024.0f) + 1e-6f);
  }
  __syncthreads();
  float rstd = red[9];

  float4 gg = *(const float4*)(g + tid * 4);
  float4 bb = *(const float4*)(bt + tid * 4);
  float4 o;
  o.x = dx * rstd * gg.x + bb.x;
  o.y = dy * rstd * gg.y + bb.y;
  o.z = dz * rstd * gg.z + bb.z;
  o.w = dw * rstd * gg.w + bb.w;
  *(float4*)(y + ((size_t)row << 10) + tid * 4) = o;
}

// ---------------------------------------------------------------------------
extern "C" void kernel_launch(void* const* d_in, const int* in_sizes, int n_in,
                              void* d_out, int out_size, void* d_ws, size_t ws_size,
                              hipStream_t stream) {
  (void)in_sizes; (void)n_in; (void)out_size; (void)ws_size;
  const float* q     = (const float*)d_in[0];
  const float* k     = (const float*)d_in[1];
  const float* v     = (const float*)d_in[2];
  const int*   mask  = (const int*)  d_in[3];
  const float* w_q   = (const float*)d_in[4];
  const float* w_k   = (const float*)d_in[5];
  const float* w_v   = (const float*)d_in[6];
  const float* w_out = (const float*)d_in[7];
  const float* ln_g  = (const float*)d_in[8];
  const float* ln_b  = (const float*)d_in[9];

  float* y    = (float*)d_out;                              // [B, QL, DM]
  float* attn = y + (size_t)B_ * QL_ * DM_;                 // [B, NH, QL, KL]

  const size_t SZ = (size_t)B_ * QL_ * DM_;                 // 4 Mi floats (16 MB)
  float* ws     = (float*)d_ws;
  float* qh     = ws;                 // [B*QL, DM]
  float* kh     = ws + SZ;            // [B*KL, DM]
  float* vh     = ws + 2 * SZ;        // [B*KL, DM]  (reused as xbuf later)
  float* concat = ws + 3 * SZ;        // [B*QL, DM]
  float* vsum   = ws + 4 * SZ;        // [B, DM]
  float* xbuf   = vh;                 // vh dead after vsum_kernel

  const int M = B_ * QL_, N = DM_, K = DM_;
  const int gemmBlocks = ((M >> 4) * (N >> 6)) / 4;         // 4 waves/block

  // Projections: qh/kh/vh = x @ W^T
  gemm16_nt_kernel<<<gemmBlocks, 128, 0, stream>>>(q, w_q, q, qh, M, N, K, 0);
  gemm16_nt_kernel<<<gemmBlocks, 128, 0, stream>>>(k, w_k, q, kh, M, N, K, 0);
  gemm16_nt_kernel<<<gemmBlocks, 128, 0, stream>>>(v, w_v, q, vh, M, N, K, 0);

  // Column sum of vh over keys (the only part of vh the reference uses)
  vsum_kernel<<<(B_ * DM_ + 255) / 256, 256, 0, stream>>>(vh, vsum);

  // Attention: scores + online softmax + attn emit + concat
  attn_kernel<<<B_ * NH_ * (QL_ / 16), 32, 0, stream>>>(qh, kh, mask, vsum,
                                                        attn, concat);

  // Out projection with residual add, then LayerNorm
  gemm16_nt_kernel<<<gemmBlocks, 128, 0, stream>>>(concat, w_out, q, xbuf, M, N, K, 1);
  ln_kernel<<<M, 256, 0, stream>>>(xbuf, ln_g, ln_b, y);
}